// CRFLoss_1726576855510
// MI455X (gfx1250) — compile-verified
//
#include <hip/hip_runtime.h>
#include <hip/hip_bf16.h>
#include <cstddef>

#define NTAG 128     // T
#define SEQ  1024    // S
#define NB   256     // B
#define AST  132     // LDS row stride (floats) -> conflict-free ds_load_b64 A-fragments

typedef float v2f __attribute__((ext_vector_type(2)));
typedef float v8f __attribute__((ext_vector_type(8)));

// Only the DEVICE pass declares amdgcn builtins; don't probe them in the host pass.
#if defined(__HIP_DEVICE_COMPILE__)
#if !__has_builtin(__builtin_amdgcn_wmma_f32_16x16x4_f32)
#error "device pass: wmma_f32_16x16x4_f32 builtin not available"
#endif
#endif

// ---------------- Kernel 0: expT = exp(transitions); copy transitions to out[1..]; zero loss
__global__ void crf_prep_kernel(const float* __restrict__ tr,
                                float* __restrict__ expT,
                                float* __restrict__ out) {
    int i = blockIdx.x * blockDim.x + threadIdx.x;
    if (i < NTAG * NTAG) {
        float t = tr[i];
        out[1 + i] = t;          // tuple output: transitions pass-through
        expT[i] = __expf(t);
    }
    if (i == 0) out[0] = 0.0f;   // loss accumulator (graph-replay safe: rewritten every call)
}

// ---------------- Kernel 1: per-batch unary + binary gather sums
__global__ void crf_scores_kernel(const float* __restrict__ inp,
                                  const unsigned char* __restrict__ masks,
                                  const int* __restrict__ tags,
                                  const float* __restrict__ tr,
                                  float* __restrict__ scores) {
    const int b = blockIdx.x;
    const int tid = threadIdx.x;
    float acc = 0.0f;
    for (int s = tid; s < SEQ; s += blockDim.x) {
        int tg = tags[b * SEQ + s];
        bool mk = masks[(size_t)b * SEQ + s] != 0;
        if (!mk) {
            acc += inp[((size_t)b * SEQ + s) * NTAG + tg];                  // unary
            if (s >= 1) acc += tr[tags[b * SEQ + s - 1] * NTAG + tg];        // binary
        }
    }
    __shared__ float red[8];
    #pragma unroll
    for (int off = 16; off >= 1; off >>= 1) acc += __shfl_xor(acc, off, 32);
    if ((tid & 31) == 0) red[tid >> 5] = acc;
    __syncthreads();
    if (tid < 8) {
        float v = red[tid];
        v += __shfl_xor(v, 4, 32);
        v += __shfl_xor(v, 2, 32);
        v += __shfl_xor(v, 1, 32);
        if (tid == 0) scores[b] = v;
    }
}

// ---------------- Kernel 2: forward scan. 16 batches / block, 8 waves = 8 j-tiles of 16.
__global__ __launch_bounds__(256, 1)
void crf_forward_kernel(const float* __restrict__ inp,
                        const unsigned char* __restrict__ masks,
                        const float* __restrict__ expT,
                        const float* __restrict__ scores,
                        float* __restrict__ out0) {
    __shared__ float sAl[16 * AST];   // alphas[16 batches][128], stride 132
    __shared__ float sP [16 * AST];   // p = exp(alpha - m)
    __shared__ float sM [16];         // per-batch running max

    const int tid  = threadIdx.x;
    const int b0   = blockIdx.x * 16;     // first batch of this group
    const int lane = tid & 31;
    const int wid  = tid >> 5;            // wave id -> j-tile
    const int lo   = lane & 15;
    const int hi   = lane >> 4;           // 0/1 half-wave
    const int n0   = wid * 16;            // tile column base

    // ---- Load B fragments of E (constant across all 1023 steps) into registers.
    // Chunk k covers K rows [4k, 4k+3]; lane layout: VGPR0={K0|K2}, VGPR1={K1|K3}.
    v2f bf[32];
    #pragma unroll
    for (int k = 0; k < 32; ++k) {
        int row = 4 * k + 2 * hi;
        int col = n0 + lo;
        bf[k].x = expT[row * NTAG + col];
        bf[k].y = expT[(row + 1) * NTAG + col];
    }

    // ---- alphas0 = inputs[:, 0, :]
    {
        int b = tid >> 4, l = tid & 15;
        const float* src = inp + ((size_t)(b0 + b) * SEQ) * NTAG + l * 8;
        float4 a = *(const float4*)src;
        float4 c = *(const float4*)(src + 4);
        *(float4*)&sAl[b * AST + l * 8]     = a;
        *(float4*)&sAl[b * AST + l * 8 + 4] = c;
    }
    __syncthreads();

    for (int t = 1; t < SEQ; ++t) {
        // ---- Phase 1: per-batch max over 128 alphas, p = exp(alpha - m).
        // thread (b = tid>>4, l = tid&15) owns 8 contiguous elements; 16-lane shfl reduce.
        {
            int b = tid >> 4, l = tid & 15;
            const float* ar = &sAl[b * AST + l * 8];
            float4 v0 = *(const float4*)ar;
            float4 v1 = *(const float4*)(ar + 4);
            float mx = fmaxf(fmaxf(fmaxf(v0.x, v0.y), fmaxf(v0.z, v0.w)),
                             fmaxf(fmaxf(v1.x, v1.y), fmaxf(v1.z, v1.w)));
            mx = fmaxf(mx, __shfl_xor(mx, 1, 32));
            mx = fmaxf(mx, __shfl_xor(mx, 2, 32));
            mx = fmaxf(mx, __shfl_xor(mx, 4, 32));
            mx = fmaxf(mx, __shfl_xor(mx, 8, 32));
            float4 p0, p1;
            p0.x = __expf(v0.x - mx); p0.y = __expf(v0.y - mx);
            p0.z = __expf(v0.z - mx); p0.w = __expf(v0.w - mx);
            p1.x = __expf(v1.x - mx); p1.y = __expf(v1.y - mx);
            p1.z = __expf(v1.z - mx); p1.w = __expf(v1.w - mx);
            *(float4*)&sP[b * AST + l * 8]     = p0;
            *(float4*)&sP[b * AST + l * 8 + 4] = p1;
            if (l == 0) sM[b] = mx;
        }
        __syncthreads();

        // ---- Phase 2: S(16x16 tile) = P(16x128) @ E(:, n0:n0+16) via 32x WMMA f32 16x16x4.
        // 4 interleaved accumulators to shorten the RAW chain on D.
        v8f acc0 = {}, acc1 = {}, acc2 = {}, acc3 = {};
        #pragma unroll
        for (int k = 0; k < 32; k += 4) {
            v2f a0 = *(const v2f*)&sP[lo * AST + 4 * (k + 0) + 2 * hi];
            acc0 = __builtin_amdgcn_wmma_f32_16x16x4_f32(false, a0, false, bf[k + 0],
                                                         (short)0, acc0, false, false);
            v2f a1 = *(const v2f*)&sP[lo * AST + 4 * (k + 1) + 2 * hi];
            acc1 = __builtin_amdgcn_wmma_f32_16x16x4_f32(false, a1, false, bf[k + 1],
                                                         (short)0, acc1, false, false);
            v2f a2 = *(const v2f*)&sP[lo * AST + 4 * (k + 2) + 2 * hi];
            acc2 = __builtin_amdgcn_wmma_f32_16x16x4_f32(false, a2, false, bf[k + 2],
                                                         (short)0, acc2, false, false);
            v2f a3 = *(const v2f*)&sP[lo * AST + 4 * (k + 3) + 2 * hi];
            acc3 = __builtin_amdgcn_wmma_f32_16x16x4_f32(false, a3, false, bf[k + 3],
                                                         (short)0, acc3, false, false);
        }
        v8f cc = (acc0 + acc1) + (acc2 + acc3);

        // ---- Phase 3: alpha[b][j] = x[b,t,j] + m_b + log(S[b][j])  (mask keeps old)
        // D layout: VGPR r -> M = r + 8*hi, N = n0 + lo.
        #pragma unroll
        for (int r = 0; r < 8; ++r) {
            int bb = r + 8 * hi;
            int j  = n0 + lo;
            size_t gidx = ((size_t)(b0 + bb) * SEQ + t) * NTAG + j;
            float xv  = inp[gidx];
            float mbv = sM[bb];
            bool  mk  = masks[(size_t)(b0 + bb) * SEQ + t] != 0;
            float nv  = xv + mbv + __logf(cc[r]);
            float ov  = sAl[bb * AST + j];
            sAl[bb * AST + j] = mk ? ov : nv;
            if (t + 1 < SEQ) __builtin_prefetch(&inp[gidx + NTAG], 0, 3);
        }
        __syncthreads();
    }

    // ---- Final: log_norm = logsumexp(alphas); loss += (log_norm - score)/B
    {
        int b = tid >> 4, l = tid & 15;
        const float* ar = &sAl[b * AST + l * 8];
        float4 v0 = *(const float4*)ar;
        float4 v1 = *(const float4*)(ar + 4);
        float mx = fmaxf(fmaxf(fmaxf(v0.x, v0.y), fmaxf(v0.z, v0.w)),
                         fmaxf(fmaxf(v1.x, v1.y), fmaxf(v1.z, v1.w)));
        mx = fmaxf(mx, __shfl_xor(mx, 1, 32));
        mx = fmaxf(mx, __shfl_xor(mx, 2, 32));
        mx = fmaxf(mx, __shfl_xor(mx, 4, 32));
        mx = fmaxf(mx, __shfl_xor(mx, 8, 32));
        float sm = __expf(v0.x - mx) + __expf(v0.y - mx) + __expf(v0.z - mx) + __expf(v0.w - mx)
                 + __expf(v1.x - mx) + __expf(v1.y - mx) + __expf(v1.z - mx) + __expf(v1.w - mx);
        sm += __shfl_xor(sm, 1, 32);
        sm += __shfl_xor(sm, 2, 32);
        sm += __shfl_xor(sm, 4, 32);
        sm += __shfl_xor(sm, 8, 32);
        if (l == 0) {
            float log_norm = mx + __logf(sm);
            float sc = scores[b0 + b];
            atomicAdd(out0, (log_norm - sc) * (1.0f / NB));
        }
    }
}

extern "C" void kernel_launch(void* const* d_in, const int* in_sizes, int n_in,
                              void* d_out, int out_size, void* d_ws, size_t ws_size,
                              hipStream_t stream) {
    const float*         inp   = (const float*)d_in[0];          // (B,S,T) f32
    const unsigned char* masks = (const unsigned char*)d_in[1];  // (B,S) bool (all false)
    const int*           tags  = (const int*)d_in[2];            // (B,S) i32
    const float*         tr    = (const float*)d_in[3];          // (T,T) f32
    float* out = (float*)d_out;                                  // [loss, transitions(16384)]

    float* expT   = (float*)d_ws;                                // 128*128 f32
    float* scores = expT + NTAG * NTAG;                          // 256 f32

    crf_prep_kernel<<<(NTAG * NTAG + 255) / 256, 256, 0, stream>>>(tr, expT, out);
    crf_scores_kernel<<<NB, 256, 0, stream>>>(inp, masks, tags, tr, scores);
    crf_forward_kernel<<<NB / 16, 256, 0, stream>>>(inp, masks, expT, scores, out);
}